// PPCNodeLayer_60017872994620
// MI455X (gfx1250) — compile-verified
//
#include <hip/hip_runtime.h>
#include <math.h>

#define NTOK 8192          // B*T = 4*2048
#define DIM  1024
#define NEXP 8
#define CAP  1024          // NTOK / NEXP
#define TSEQ 2048
#define NELEM (NTOK * DIM) // 8388608

typedef __attribute__((ext_vector_type(2))) float v2f;
typedef __attribute__((ext_vector_type(8))) float v8f;

struct Ctrl {
    unsigned long long ressum;  // fixed-point |residual| sum (deterministic)
    float lr;
    int gate;       // 1 => loop finished (done or max iters)
    int iters;
    int maxIters;
};

#define RES_SCALE 1048576.0f   // 2^20 fixed point

// ---------------------------------------------------------------- init
__global__ void init_kernel(const float* __restrict__ xr, const float* __restrict__ xi,
                            float* __restrict__ zr, float* __restrict__ zi,
                            const int* __restrict__ li, Ctrl* ctrl) {
    int m = blockIdx.x * blockDim.x + threadIdx.x;
    if (m < NELEM) { zr[m] = xr[m]; zi[m] = xi[m]; }
    if (m == 0) {
        ctrl->gate = 0;
        ctrl->iters = 0;
        int mi = li[0] - 1; if (mi < 1) mi = 1;
        ctrl->maxIters = mi;
        ctrl->lr = 0.5f;
        ctrl->ressum = 0ull;
    }
}

// -------------------------------------------------- rotation tables
__global__ void rot_kernel(const float* __restrict__ phase,
                           float* __restrict__ rotc, float* __restrict__ rots) {
    int d = blockIdx.x * blockDim.x + threadIdx.x;
    if (d < DIM) {
        float s, c;
        sincosf(phase[d], &s, &c);
        rotc[d] = c;
        rots[d] = s;
    }
}

// ------------------------------------------------------- rank reset
__global__ void prep_kernel(int* __restrict__ rank, Ctrl* ctrl, int gated) {
    if (gated && ctrl->gate) return;
    int m = blockIdx.x * blockDim.x + threadIdx.x;
    if (m < NEXP * NTOK) rank[m] = -1;
    if (m == 0) ctrl->ressum = 0ull;
}

// ---------------------------------------------------------- gating
// One wave per token: logits = z.real @ Wg + bg ; softmax ; write S^T [E][NTOK]
__global__ void gating_kernel(const float* __restrict__ zr, const float* __restrict__ Wg,
                              const float* __restrict__ bg, float* __restrict__ ST,
                              const Ctrl* __restrict__ ctrl, int gated) {
    if (gated && ctrl->gate) return;
    int lane = threadIdx.x & 31;
    int wv   = threadIdx.x >> 5;
    int n    = blockIdx.x * 8 + wv;
    float acc[NEXP];
#pragma unroll
    for (int e = 0; e < NEXP; ++e) acc[e] = 0.f;
    const float* zrow = zr + (size_t)n * DIM;
    for (int d = lane; d < DIM; d += 32) {
        float zv = zrow[d];
        const float* wrow = Wg + d * NEXP;
#pragma unroll
        for (int e = 0; e < NEXP; ++e) acc[e] += zv * wrow[e];
    }
#pragma unroll
    for (int off = 16; off > 0; off >>= 1) {
#pragma unroll
        for (int e = 0; e < NEXP; ++e) acc[e] += __shfl_down(acc[e], off, 32);
    }
    if (lane == 0) {
        float mx = -1e30f;
#pragma unroll
        for (int e = 0; e < NEXP; ++e) { acc[e] += bg[e]; mx = fmaxf(mx, acc[e]); }
        float s = 0.f;
#pragma unroll
        for (int e = 0; e < NEXP; ++e) { acc[e] = expf(acc[e] - mx); s += acc[e]; }
        float inv = 1.f / s;
#pragma unroll
        for (int e = 0; e < NEXP; ++e) ST[e * NTOK + n] = acc[e] * inv;
    }
}

// ------------------------------------------------------------ top-k
// One block per expert; bitonic sort (descending) of 8192 u64 keys in LDS.
__global__ void __launch_bounds__(1024)
topk_kernel(const float* __restrict__ ST, float* __restrict__ scoresTop,
            int* __restrict__ idxTop, const Ctrl* __restrict__ ctrl, int gated) {
    if (gated && ctrl->gate) return;
    __shared__ unsigned long long key[NTOK];   // 64 KB (WGP has 320 KB LDS)
    const int e   = blockIdx.x;
    const int tid = threadIdx.x;
    for (int i = tid; i < NTOK; i += 1024) {
        unsigned u = __float_as_uint(ST[e * NTOK + i]);   // scores > 0 -> monotone bits
        key[i] = ((unsigned long long)u << 32) | (unsigned)i;
    }
    __syncthreads();
    for (int k = 2; k <= NTOK; k <<= 1) {
        for (int j = k >> 1; j > 0; j >>= 1) {
            for (int i = tid; i < NTOK; i += 1024) {
                int ij = i ^ j;
                if (ij > i) {
                    bool up = ((i & k) == 0);   // descending overall
                    unsigned long long a = key[i], b = key[ij];
                    if (up ? (a < b) : (a > b)) { key[i] = b; key[ij] = a; }
                }
            }
            __syncthreads();
        }
    }
    if (tid < CAP) {
        unsigned long long kk = key[tid];
        scoresTop[e * CAP + tid] = __uint_as_float((unsigned)(kk >> 32));
        idxTop[e * CAP + tid]    = (int)(kk & 0xFFFFFFFFu);
    }
}

// ------------------------------------------------------- rank scatter
__global__ void scatter_kernel(const int* __restrict__ idxTop, int* __restrict__ rank,
                               const Ctrl* __restrict__ ctrl, int gated) {
    if (gated && ctrl->gate) return;
    int g = blockIdx.x * blockDim.x + threadIdx.x;   // [0, NEXP*CAP)
    int e = g >> 10, k = g & (CAP - 1);
    rank[e * NTOK + idxTop[e * CAP + k]] = k;
}

// ----------------------------------------------------- expert GEMM (WMMA f32)
// out[e][k][n] (real & imag) = gathered[e][k][:] @ We[e][:,n]
// Wave: 32x32 tile (2 M-subtiles x 2 N-subtiles x {re,im} = 8 accumulators),
// so each A fragment feeds 2 WMMAs and each B fragment feeds 4 -> 1 VMEM/WMMA.
__global__ void __launch_bounds__(256)
gemm_kernel(const float* __restrict__ zr, const float* __restrict__ zi,
            const float* __restrict__ We, const int* __restrict__ idxTop,
            float* __restrict__ outr, float* __restrict__ outi,
            const Ctrl* __restrict__ ctrl, int gated) {
    if (gated && ctrl->gate) return;
    const int lane = threadIdx.x & 31;
    const int wv   = threadIdx.x >> 5;
    const int e    = blockIdx.z;
    const int m0   = blockIdx.y << 5;                    // 32 rows per wave
    const int n0   = (blockIdx.x << 8) + (wv << 5);      // 32 cols per wave
    const int half = lane >> 4;          // 0: K=k..k+1, 1: K=k+2..k+3
    const int l16  = lane & 15;
    const int tok0 = idxTop[e * CAP + m0 + l16];
    const int tok1 = idxTop[e * CAP + m0 + 16 + l16];
    const float* Ar0 = zr + (size_t)tok0 * DIM;
    const float* Ai0 = zi + (size_t)tok0 * DIM;
    const float* Ar1 = zr + (size_t)tok1 * DIM;
    const float* Ai1 = zi + (size_t)tok1 * DIM;
    const float* B0  = We + (size_t)e * DIM * DIM + n0 + l16;
    const float* B1  = B0 + 16;
    v8f a00r = {}, a00i = {}, a01r = {}, a01i = {};
    v8f a10r = {}, a10i = {}, a11r = {}, a11i = {};
#pragma unroll 2
    for (int k = 0; k < DIM; k += 4) {
        int kl = k + half * 2;
        v2f f0r = *(const v2f*)(Ar0 + kl);
        v2f f0i = *(const v2f*)(Ai0 + kl);
        v2f f1r = *(const v2f*)(Ar1 + kl);
        v2f f1i = *(const v2f*)(Ai1 + kl);
        v2f b0, b1;
        b0.x = B0[(size_t)kl * DIM];
        b0.y = B0[(size_t)(kl + 1) * DIM];
        b1.x = B1[(size_t)kl * DIM];
        b1.y = B1[(size_t)(kl + 1) * DIM];
        a00r = __builtin_amdgcn_wmma_f32_16x16x4_f32(false, f0r, false, b0, (short)0, a00r, false, false);
        a00i = __builtin_amdgcn_wmma_f32_16x16x4_f32(false, f0i, false, b0, (short)0, a00i, false, false);
        a01r = __builtin_amdgcn_wmma_f32_16x16x4_f32(false, f0r, false, b1, (short)0, a01r, false, false);
        a01i = __builtin_amdgcn_wmma_f32_16x16x4_f32(false, f0i, false, b1, (short)0, a01i, false, false);
        a10r = __builtin_amdgcn_wmma_f32_16x16x4_f32(false, f1r, false, b0, (short)0, a10r, false, false);
        a10i = __builtin_amdgcn_wmma_f32_16x16x4_f32(false, f1i, false, b0, (short)0, a10i, false, false);
        a11r = __builtin_amdgcn_wmma_f32_16x16x4_f32(false, f1r, false, b1, (short)0, a11r, false, false);
        a11i = __builtin_amdgcn_wmma_f32_16x16x4_f32(false, f1i, false, b1, (short)0, a11i, false, false);
    }
    const int n_lo = n0 + l16;
#pragma unroll
    for (int j = 0; j < 8; ++j) {
        int mA = m0 + j + half * 8;          // M-subtile 0
        int mB = mA + 16;                    // M-subtile 1
        size_t oA = ((size_t)(e * CAP + mA)) * DIM + n_lo;
        size_t oB = ((size_t)(e * CAP + mB)) * DIM + n_lo;
        outr[oA]      = a00r[j];
        outi[oA]      = a00i[j];
        outr[oA + 16] = a01r[j];
        outi[oA + 16] = a01i[j];
        outr[oB]      = a10r[j];
        outi[oB]      = a10i[j];
        outr[oB + 16] = a11r[j];
        outi[oB + 16] = a11i[j];
    }
}

// ----------------------------------------------- target & pred helpers
__device__ __forceinline__ void target_at(const float* __restrict__ xr,
                                          const float* __restrict__ xi,
                                          const float* __restrict__ rotc,
                                          const float* __restrict__ rots,
                                          int m, float& tr, float& ti) {
    int n = m >> 10;
    int d = m & (DIM - 1);
    int t = n & (TSEQ - 1);
    if (t == 0) { tr = xr[m]; ti = xi[m]; }
    else {
        float c = rotc[d], s = rots[d];
        float pr = xr[m - DIM], pim = xi[m - DIM];
        tr = pr * c - pim * s;
        ti = pr * s + pim * c;
    }
}

__device__ __forceinline__ void pred_at(const int* __restrict__ rank,
                                        const float* __restrict__ scores,
                                        const float* __restrict__ outr,
                                        const float* __restrict__ outi,
                                        const float* __restrict__ be,
                                        int n, int d, float& pr, float& pim) {
    pr = 0.f; pim = 0.f;
#pragma unroll
    for (int e = 0; e < NEXP; ++e) {      // fixed order -> deterministic
        int rk = rank[e * NTOK + n];
        if (rk >= 0) {
            float s = scores[e * CAP + rk];
            size_t o = ((size_t)(e * CAP + rk)) * DIM + d;
            pr  += s * (outr[o] + be[e * DIM + d]);
            pim += s * outi[o];
        }
    }
}

// ------------------------------------------------- residual + z update
__global__ void resupdate_kernel(const float* __restrict__ xr, const float* __restrict__ xi,
                                 const float* __restrict__ rotc, const float* __restrict__ rots,
                                 const int* __restrict__ rank, const float* __restrict__ scores,
                                 const float* __restrict__ outr, const float* __restrict__ outi,
                                 const float* __restrict__ be,
                                 float* __restrict__ zr, float* __restrict__ zi, Ctrl* ctrl) {
    if (ctrl->gate) return;
    __shared__ float red[256];
    int m = blockIdx.x * 256 + threadIdx.x;
    int n = m >> 10, d = m & (DIM - 1);
    float tr, ti, pr, pim;
    target_at(xr, xi, rotc, rots, m, tr, ti);
    pred_at(rank, scores, outr, outi, be, n, d, pr, pim);
    float rr = tr - pr, ri = ti - pim;
    float lr = ctrl->lr;
    zr[m] += lr * rr;
    zi[m] += lr * ri;
    red[threadIdx.x] = sqrtf(rr * rr + ri * ri);
    __syncthreads();
    for (int s = 128; s > 0; s >>= 1) {
        if (threadIdx.x < s) red[threadIdx.x] += red[threadIdx.x + s];
        __syncthreads();
    }
    if (threadIdx.x == 0) {
        unsigned long long q = (unsigned long long)(red[0] * RES_SCALE);
        atomicAdd(&ctrl->ressum, q);      // integer: order-independent
    }
}

// ---------------------------------------------------------- finalize
__global__ void finalize_kernel(Ctrl* ctrl) {
    if (ctrl->gate) return;
    double mean = (double)ctrl->ressum / (double)RES_SCALE / (double)NELEM;
    ctrl->iters += 1;
    ctrl->lr *= 0.85f;
    if (mean < 0.001) ctrl->gate = 1;
    if (ctrl->iters >= ctrl->maxIters) ctrl->gate = 1;
}

// ------------------------------------------------------------- output
__global__ void final_kernel(const float* __restrict__ xr, const float* __restrict__ xi,
                             const float* __restrict__ rotc, const float* __restrict__ rots,
                             const int* __restrict__ rank, const float* __restrict__ scores,
                             const float* __restrict__ outr, const float* __restrict__ outi,
                             const float* __restrict__ be,
                             const float* __restrict__ zr, const float* __restrict__ zi,
                             float* __restrict__ out, const Ctrl* __restrict__ ctrl) {
    int m = blockIdx.x * 256 + threadIdx.x;
    int n = m >> 10, d = m & (DIM - 1);
    float tr, ti, pr, pim;
    target_at(xr, xi, rotc, rots, m, tr, ti);
    pred_at(rank, scores, outr, outi, be, n, d, pr, pim);
    size_t o = (size_t)m * 2;
    out[o]     = zr[m] + 0.5f * (tr - pr);
    out[o + 1] = zi[m] + 0.5f * (ti - pim);
    if (m == 0) ((int*)out)[(size_t)NELEM * 2] = ctrl->iters;
}

// ------------------------------------------------------------- launch
extern "C" void kernel_launch(void* const* d_in, const int* in_sizes, int n_in,
                              void* d_out, int out_size, void* d_ws, size_t ws_size,
                              hipStream_t stream) {
    const float* xr    = (const float*)d_in[0];
    const float* xi    = (const float*)d_in[1];
    const float* phase = (const float*)d_in[2];
    const float* Wg    = (const float*)d_in[3];
    const float* bg    = (const float*)d_in[4];
    const float* We    = (const float*)d_in[5];
    const float* be    = (const float*)d_in[6];
    const int*   li    = (const int*)d_in[7];

    char* ws = (char*)d_ws;
    Ctrl* ctrl = (Ctrl*)ws;
    size_t off = 256;
    float* zr        = (float*)(ws + off); off += (size_t)NELEM * 4;
    float* zi        = (float*)(ws + off); off += (size_t)NELEM * 4;
    float* outr      = (float*)(ws + off); off += (size_t)NEXP * CAP * DIM * 4;
    float* outi      = (float*)(ws + off); off += (size_t)NEXP * CAP * DIM * 4;
    float* ST        = (float*)(ws + off); off += (size_t)NEXP * NTOK * 4;
    float* scoresTop = (float*)(ws + off); off += (size_t)NEXP * CAP * 4;
    int*   idxTop    = (int*)  (ws + off); off += (size_t)NEXP * CAP * 4;
    int*   rank      = (int*)  (ws + off); off += (size_t)NEXP * NTOK * 4;
    float* rotc      = (float*)(ws + off); off += (size_t)DIM * 4;
    float* rots      = (float*)(ws + off); off += (size_t)DIM * 4;

    dim3 b256(256);
    init_kernel<<<NELEM / 256, b256, 0, stream>>>(xr, xi, zr, zi, li, ctrl);
    rot_kernel<<<DIM / 256, b256, 0, stream>>>(phase, rotc, rots);

    auto moe = [&](int gated) {
        prep_kernel<<<(NEXP * NTOK) / 256, b256, 0, stream>>>(rank, ctrl, gated);
        gating_kernel<<<NTOK / 8, b256, 0, stream>>>(zr, Wg, bg, ST, ctrl, gated);
        topk_kernel<<<NEXP, dim3(1024), 0, stream>>>(ST, scoresTop, idxTop, ctrl, gated);
        scatter_kernel<<<(NEXP * CAP) / 256, b256, 0, stream>>>(idxTop, rank, ctrl, gated);
        gemm_kernel<<<dim3(DIM / 256, CAP / 32, NEXP), b256, 0, stream>>>(
            zr, zi, We, idxTop, outr, outi, ctrl, gated);
    };

    for (int it = 0; it < 7; ++it) {       // max_iters for local_iterations=8
        moe(1);
        resupdate_kernel<<<NELEM / 256, b256, 0, stream>>>(
            xr, xi, rotc, rots, rank, scoresTop, outr, outi, be, zr, zi, ctrl);
        finalize_kernel<<<1, 1, 0, stream>>>(ctrl);
    }
    moe(0);                                // final pred at z_final
    final_kernel<<<NELEM / 256, b256, 0, stream>>>(
        xr, xi, rotc, rots, rank, scoresTop, outr, outi, be, zr, zi, (float*)d_out, ctrl);
}